// QuantumTextEncoder_6957847020218
// MI455X (gfx1250) — compile-verified
//
#include <hip/hip_runtime.h>
#include <stdint.h>

#define DD     768
#define QQ     10
#define NSTATE 1024
#define BLOCK  320          // 10 wave32s: one wave per qubit in the chain phase
#define CHUNK  24           // 768 / 32 lanes
#define PI_F   3.14159265358979323846f

typedef unsigned int v4u __attribute__((ext_vector_type(4)));
typedef int          v8i __attribute__((ext_vector_type(8)));
typedef int          v4i __attribute__((ext_vector_type(4)));
typedef float        v2f __attribute__((ext_vector_type(2)));
typedef float        v8f __attribute__((ext_vector_type(8)));

__device__ __forceinline__ float2 cmul(float2 a, float2 b) {
    return make_float2(__builtin_fmaf(a.x, b.x, -(a.y * b.y)),
                       __builtin_fmaf(a.x, b.y,  (a.y * b.x)));
}
__device__ __forceinline__ float2 cadd(float2 a, float2 b) {
    return make_float2(a.x + b.x, a.y + b.y);
}
__device__ __forceinline__ float2 shfl2(float2 v, int off) {
    return make_float2(__shfl_down(v.x, (unsigned)off, 32),
                       __shfl_down(v.y, (unsigned)off, 32));
}

__global__ __launch_bounds__(BLOCK) void qenc_kernel(
    const float* __restrict__ emb,   // [B*S, 768]
    const float* __restrict__ prm,   // [768, 10, 3]
    float2* __restrict__ out)        // [B*S, 1024] complex64
{
    __shared__ float  scaled[DD];    // 0.5*tanh(emb)*pi, staged via TDM
    __shared__ float2 vq[QQ][2];     // per-qubit final 2-vector

    const int tid  = threadIdx.x;
    const int bs   = blockIdx.x;
    const int w    = tid >> 5;
    const int lane = tid & 31;

    // ---- Stage embedding row (3 KB) into LDS via Tensor Data Mover ----
#if defined(__gfx1250__) && __has_builtin(__builtin_amdgcn_tensor_load_to_lds) && \
    __has_builtin(__builtin_amdgcn_s_wait_tensorcnt)
    if (w == 0) {   // one wave issues the TDM descriptor (EXEC ignored by TDM)
        uint32_t lds_off = (uint32_t)(uintptr_t)&scaled[0];   // low 32b of LDS generic addr
        uint64_t ga      = (uint64_t)(uintptr_t)(emb + (size_t)bs * DD);

        v4u g0;
        g0[0] = 1u;                                  // count=1 (valid), user mode
        g0[1] = lds_off;                             // lds_addr (bytes)
        g0[2] = (uint32_t)(ga & 0xFFFFFFFFu);        // global_addr[31:0]
        g0[3] = (uint32_t)((ga >> 32) & 0x01FFFFFFu) // global_addr[56:32]
              | (2u << 30);                          // type = 2 ("image")

        v8i g1;
        g1[0] = (int)(2u << 16);                     // data_size=2 (4B)
        g1[1] = (int)((DD & 0xFFFFu) << 16);         // tensor_dim0[15:0]
        g1[2] = (int)(((DD >> 16) & 0xFFFFu) | (1u << 16)); // dim0 hi | tensor_dim1=1
        g1[3] = (int)((DD & 0xFFFFu) << 16);         // dim1 hi | tile_dim0=768
        g1[4] = 0;                                   // tile_dim1=0, tile_dim2=0
        g1[5] = DD;                                  // tensor_dim0_stride lo
        g1[6] = (int)((DD & 0xFFFFu) << 16);         // stride0 hi | stride1 lo
        g1[7] = 0;                                   // stride1 hi

        v4i z4 = {0, 0, 0, 0};
#if defined(__clang_major__) && (__clang_major__ >= 23)
        v8i z8 = {0, 0, 0, 0, 0, 0, 0, 0};
        __builtin_amdgcn_tensor_load_to_lds(g0, g1, z4, z4, z8, 0);
#else
        __builtin_amdgcn_tensor_load_to_lds(g0, g1, z4, z4, 0);
#endif
        __builtin_amdgcn_s_wait_tensorcnt(0);
    }
#else
    for (int i = tid; i < DD; i += BLOCK) scaled[i] = emb[(size_t)bs * DD + i];
#endif
    __syncthreads();

    // scaled = 0.5 * tanh(emb) * pi  (in place; folds the half-angle factor)
    for (int i = tid; i < DD; i += BLOCK) scaled[i] = tanhf(scaled[i]) * (0.5f * PI_F);
    __syncthreads();

    // ---- Ordered product of 768 composite gates, split 24 per lane ----
    {
        const int q = w;             // one wave per qubit
        float2 m00 = {1.f, 0.f}, m01 = {0.f, 0.f};
        float2 m10 = {0.f, 0.f}, m11 = {1.f, 0.f};

        const int d0 = lane * CHUNK;
        for (int k = 0; k < CHUNK; ++k) {
            const int d = d0 + k;
            const float h = scaled[d];
            const float* p = prm + ((size_t)d * QQ + q) * 3;
            const float a0 = h * p[0], a1 = h * p[1], a2 = h * p[2];

            float c0, s0, c1, s1, c2, s2;
            sincosf(a0, &s0, &c0);
            sincosf(a1, &s1, &c1);
            sincosf(a2, &s2, &c2);

            const float2 ph  = {c2, -s2};        // exp(-i*a2)
            const float2 phc = {c2,  s2};        // conj
            const float2 g00 = cmul(ph,  make_float2( c1 * c0,  s1 * s0));
            const float2 g01 = cmul(ph,  make_float2(-s1 * c0, -c1 * s0));
            const float2 g10 = cmul(phc, make_float2( s1 * c0, -c1 * s0));
            const float2 g11 = cmul(phc, make_float2( c1 * c0, -s1 * s0));

            // M <- G * M   (later gates multiply on the left)
            const float2 n00 = cadd(cmul(g00, m00), cmul(g01, m10));
            const float2 n01 = cadd(cmul(g00, m01), cmul(g01, m11));
            const float2 n10 = cadd(cmul(g10, m00), cmul(g11, m10));
            const float2 n11 = cadd(cmul(g10, m01), cmul(g11, m11));
            m00 = n00; m01 = n01; m10 = n10; m11 = n11;
        }

        // log2(32) tree combine across lanes: partner (higher d) on the left
#pragma unroll
        for (int off = 1; off < 32; off <<= 1) {
            const float2 o00 = shfl2(m00, off), o01 = shfl2(m01, off);
            const float2 o10 = shfl2(m10, off), o11 = shfl2(m11, off);
            const float2 n00 = cadd(cmul(o00, m00), cmul(o01, m10));
            const float2 n01 = cadd(cmul(o00, m01), cmul(o01, m11));
            const float2 n10 = cadd(cmul(o10, m00), cmul(o11, m10));
            const float2 n11 = cadd(cmul(o10, m01), cmul(o11, m11));
            m00 = n00; m01 = n01; m10 = n10; m11 = n11;
        }

        if (lane == 0) {             // v = M @ (1,0)^T -> first column
            vq[q][0] = m00;
            vq[q][1] = m10;
        }
    }
    __syncthreads();

    // ---- Kron expansion as 32x32 complex outer product: state = sA (x) sB ----
    // sA[i] = prod_{q=0..4} vq[q][bit(4-q) of i],  sB[j] = prod_{q=5..9} vq[q][bit(9-q) of j]
    // state[i*32+j] = sA[i]*sB[j]: Re = aRe@bRe - aIm@bIm, Im = aIm@bRe + aRe@bIm
    // -> two V_WMMA_F32_16X16X4_F32 per 16x16 block sharing the B operand.
#if defined(__gfx1250__) && __has_builtin(__builtin_amdgcn_wmma_f32_16x16x4_f32)
    if (w < 8) {                     // 8 waves: {Re,Im} x {mi} x {ni}
        const int comp = (w >> 2) & 1;   // 0 = Re, 1 = Im
        const int mi   =  w       & 1;   // sA 16-row block
        const int ni   = (w >> 1) & 1;   // sB 16-col block
        const int L    = lane & 15;
        const bool lo  = lane < 16;

        // per-lane half-kron partial products (4 cmuls each)
        const int i = mi * 16 + L;
        float2 a = vq[0][(i >> 4) & 1];
#pragma unroll
        for (int q = 1; q < 5; ++q) a = cmul(a, vq[q][(i >> (4 - q)) & 1]);

        const int j = ni * 16 + L;
        float2 b = vq[5][(j >> 4) & 1];
#pragma unroll
        for (int q = 6; q < 10; ++q) b = cmul(b, vq[q][(j >> (9 - q)) & 1]);

        // A 16x4: lanes 0-15 = K0/K1, lanes 16-31 = K2/K3 (zeroed). Same for B.
        v2f A, B;
        if (lo) {
            A[0] = comp ? a.y : a.x;          // K0 column
            A[1] = comp ? a.x : -a.y;         // K1 column
            B[0] = b.x;                       // K0 row
            B[1] = b.y;                       // K1 row
        } else {
            A[0] = 0.f; A[1] = 0.f; B[0] = 0.f; B[1] = 0.f;
        }

        v8f c = {0.f, 0.f, 0.f, 0.f, 0.f, 0.f, 0.f, 0.f};
        v8f d = __builtin_amdgcn_wmma_f32_16x16x4_f32(
            /*neg_a=*/false, A, /*neg_b=*/false, B,
            /*c_mod=*/(short)0, c, /*reuse_a=*/false, /*reuse_b=*/false);

        // D 16x16 f32: VGPR r -> M = r (lanes 0-15) / M = 8+r (lanes 16-31), N = lane&15
        float* outf = (float*)(out + (size_t)bs * NSTATE);
#pragma unroll
        for (int r = 0; r < 8; ++r) {
            const int m = r + (lo ? 0 : 8);
            const int n = (mi * 16 + m) * 32 + (ni * 16 + L);
            outf[2 * n + comp] = d[r];
        }
    }
#else
    for (int n = tid; n < NSTATE; n += BLOCK) {
        float2 acc = vq[0][(n >> 9) & 1];
#pragma unroll
        for (int qq = 1; qq < QQ; ++qq)
            acc = cmul(acc, vq[qq][(n >> (9 - qq)) & 1]);
        out[(size_t)bs * NSTATE + n] = acc;
    }
#endif
}

extern "C" void kernel_launch(void* const* d_in, const int* in_sizes, int n_in,
                              void* d_out, int out_size, void* d_ws, size_t ws_size,
                              hipStream_t stream) {
    (void)n_in; (void)out_size; (void)d_ws; (void)ws_size;
    const float* emb = (const float*)d_in[0];   // [B,S,768] f32
    const float* prm = (const float*)d_in[1];   // [768,10,3] f32
    float2* out = (float2*)d_out;               // complex64 as float2
    const int bsCount = in_sizes[0] / DD;       // B*S = 512
    qenc_kernel<<<bsCount, BLOCK, 0, stream>>>(emb, prm, out);
}